// LSTM_16458314678670
// MI455X (gfx1250) — compile-verified
//
#include <hip/hip_runtime.h>

typedef __attribute__((ext_vector_type(2))) float v2f;
typedef __attribute__((ext_vector_type(8))) float v8f;

#define T_LEN 262144
#define HID   32
#define G4    128   // 4*H
#define CHUNK 4     // scan software-pipeline depth

// ---------------------------------------------------------------------------
// Helpers
// ---------------------------------------------------------------------------
__device__ __forceinline__ float bcast_lane(float v, int j) {
  // wave32 broadcast of lane j -> SGPR; feeds FMA as a scalar operand
  return __int_as_float(__builtin_amdgcn_readlane(__float_as_int(v), j));
}
__device__ __forceinline__ float sigm(float v) {
  return 1.0f / (1.0f + __expf(-v));
}
__device__ __forceinline__ float tanh_fast(float v) {
  v = fminf(15.0f, fmaxf(-15.0f, v));
  float e = __expf(2.0f * v);
  return (e - 1.0f) / (e + 1.0f);
}

// ---------------------------------------------------------------------------
// Phase 1: gates_x[t][g] = sum_k x[t][k] * W_ih[g][k] + b_ih[g] + b_hh[g]
// One V_WMMA_F32_16X16X4_F32 per 16(t) x 16(g) tile. K = I = 4 exactly.
// Block = 256 threads = 8 waves; wave w handles gate-column tile w (8*16=128).
// ---------------------------------------------------------------------------
__global__ __launch_bounds__(256)
void lstm_gates_x_wmma(const float* __restrict__ x,
                       const float* __restrict__ W_ih,
                       const float* __restrict__ b_ih,
                       const float* __restrict__ b_hh,
                       float* __restrict__ gx) {
  const int wave = threadIdx.x >> 5;
  const int lane = threadIdx.x & 31;
  const int t0   = blockIdx.x * 16;
  const int g0   = wave * 16;
  const int half = lane >> 4;      // 0: K 0..1, 1: K 2..3
  const int l15  = lane & 15;

  // A (16x4, f32): lane<16 holds row t0+l15 @ K=0,1 ; lane>=16 @ K=2,3
  v2f a;
  { const float* ap = x + (size_t)(t0 + l15) * 4 + 2 * half;
    a.x = ap[0]; a.y = ap[1]; }

  // B (4x16, f32): B[k][n] = W_ih[g0+n][k]; W_ih rows are 4 contiguous floats
  v2f b;
  { const float* bp = W_ih + (size_t)(g0 + l15) * 4 + 2 * half;
    b.x = bp[0]; b.y = bp[1]; }

  // C init = bias (independent of row m)
  float bias = b_ih[g0 + l15] + b_hh[g0 + l15];
  v8f c;
#pragma unroll
  for (int r = 0; r < 8; ++r) c[r] = bias;

  // D = A x B + C   (8 args: neg_a, A, neg_b, B, c_mod, C, reuse_a, reuse_b)
  c = __builtin_amdgcn_wmma_f32_16x16x4_f32(false, a, false, b,
                                            (short)0, c, false, false);

  // D layout: VGPR r -> M = r + 8*half, N = l15
#pragma unroll
  for (int r = 0; r < 8; ++r) {
    int m = r + 8 * half;
    gx[(size_t)(t0 + m) * G4 + g0 + l15] = c[r];
  }
}

// ---------------------------------------------------------------------------
// Phase 2: sequential LSTM scan, ONE wave32.
// Lane l owns gate rows {l, 32+l, 64+l, 96+l} of W_hh in registers (128 VGPRs).
// h is distributed one element per lane; broadcasts use v_readlane.
// gates_x rows are register double-buffered CHUNK steps ahead to hide L2 latency.
// ---------------------------------------------------------------------------
template<bool FUSED>
__global__ __launch_bounds__(32)
void lstm_scan(const float* __restrict__ gx,     // used if !FUSED
               const float* __restrict__ x,      // used if FUSED
               const float* __restrict__ W_ih,   // used if FUSED
               const float* __restrict__ W_hh,
               const float* __restrict__ b_ih,
               const float* __restrict__ b_hh,
               const float* __restrict__ fc_w,
               const float* __restrict__ fc_b,
               float* __restrict__ out) {
  const int lane = threadIdx.x & 31;

  // Recurrent weights, register-resident (single-wave kernel: occupancy moot)
  float w0[HID], w1[HID], w2[HID], w3[HID];
#pragma unroll
  for (int j = 0; j < HID; ++j) {
    w0[j] = W_hh[(size_t)(0 * HID + lane) * HID + j];
    w1[j] = W_hh[(size_t)(1 * HID + lane) * HID + j];
    w2[j] = W_hh[(size_t)(2 * HID + lane) * HID + j];
    w3[j] = W_hh[(size_t)(3 * HID + lane) * HID + j];
  }

  // Fused path: input weights + biases per lane
  float bi[4] = {0.f, 0.f, 0.f, 0.f};
  float wi[4][4];
  if (FUSED) {
#pragma unroll
    for (int g = 0; g < 4; ++g) {
      bi[g] = b_ih[g * HID + lane] + b_hh[g * HID + lane];
#pragma unroll
      for (int k = 0; k < 4; ++k)
        wi[g][k] = W_ih[(size_t)(g * HID + lane) * 4 + k];
    }
  }

  float h = 0.0f, cst = 0.0f;
  float cur[CHUNK][4], nxt[CHUNK][4];

  auto load_chunk = [&](float dst[CHUNK][4], int tbase) {
#pragma unroll
    for (int s = 0; s < CHUNK; ++s) {
      if (!FUSED) {
        const float* p = gx + (size_t)(tbase + s) * G4 + lane;
        dst[s][0] = p[0 * HID];
        dst[s][1] = p[1 * HID];
        dst[s][2] = p[2 * HID];
        dst[s][3] = p[3 * HID];
      } else {
        const float* xp = x + (size_t)(tbase + s) * 4;  // uniform -> s_load
        float x0 = xp[0], x1 = xp[1], x2 = xp[2], x3 = xp[3];
#pragma unroll
        for (int g = 0; g < 4; ++g)
          dst[s][g] = bi[g] + x0 * wi[g][0] + x1 * wi[g][1]
                            + x2 * wi[g][2] + x3 * wi[g][3];
      }
    }
  };

  load_chunk(cur, 0);

  for (int t0 = 0; t0 < T_LEN; t0 += CHUNK) {
    if (t0 + CHUNK < T_LEN) load_chunk(nxt, t0 + CHUNK);  // prefetch ahead

#pragma unroll
    for (int s = 0; s < CHUNK; ++s) {
      float a0 = cur[s][0], a1 = cur[s][1], a2 = cur[s][2], a3 = cur[s][3];
#pragma unroll
      for (int j = 0; j < HID; ++j) {
        float hj = bcast_lane(h, j);         // v_readlane -> scalar FMA operand
        a0 += hj * w0[j];
        a1 += hj * w1[j];
        a2 += hj * w2[j];
        a3 += hj * w3[j];
      }
      float ig = sigm(a0);
      float fg = sigm(a1);
      float gg = tanh_fast(a2);
      float og = sigm(a3);
      cst = fg * cst + ig * gg;
      h   = og * tanh_fast(cst);
    }

#pragma unroll
    for (int s = 0; s < CHUNK; ++s)
#pragma unroll
      for (int g = 0; g < 4; ++g)
        cur[s][g] = nxt[s][g];
  }

  // out = h . fc_w + fc_b  (O = 1); wave32 xor-shuffle reduction
  float p = h * fc_w[lane];
#pragma unroll
  for (int off = 16; off > 0; off >>= 1)
    p += __shfl_xor(p, off, 32);
  if (lane == 0) out[0] = p + fc_b[0];
}

// ---------------------------------------------------------------------------
// Host launcher
// ---------------------------------------------------------------------------
extern "C" void kernel_launch(void* const* d_in, const int* in_sizes, int n_in,
                              void* d_out, int out_size, void* d_ws, size_t ws_size,
                              hipStream_t stream) {
  (void)in_sizes; (void)n_in; (void)out_size;
  const float* x    = (const float*)d_in[0];
  const float* W_ih = (const float*)d_in[1];
  const float* W_hh = (const float*)d_in[2];
  const float* b_ih = (const float*)d_in[3];
  const float* b_hh = (const float*)d_in[4];
  const float* fc_w = (const float*)d_in[5];
  const float* fc_b = (const float*)d_in[6];
  float* out = (float*)d_out;

  const size_t need = (size_t)T_LEN * G4 * sizeof(float);  // 128 MiB
  if (ws_size >= need) {
    float* gx = (float*)d_ws;
    lstm_gates_x_wmma<<<T_LEN / 16, 256, 0, stream>>>(x, W_ih, b_ih, b_hh, gx);
    lstm_scan<false><<<1, 32, 0, stream>>>(gx, x, W_ih, W_hh, b_ih, b_hh,
                                           fc_w, fc_b, out);
  } else {
    lstm_scan<true><<<1, 32, 0, stream>>>(nullptr, x, W_ih, W_hh, b_ih, b_hh,
                                          fc_w, fc_b, out);
  }
}